// ArticulatoryMetricLoss_23407571763518
// MI455X (gfx1250) — compile-verified
//
#include <hip/hip_runtime.h>

// Problem constants (match reference).
#define B_SZ 4096
#define D_SZ 768
#define MT 4                       // 4x4 WMMA tiles per wave
#define MACRO (MT * 16)            // 64x64 macro tile per wave
#define WPB 8                      // waves per block (256 threads, wave32)
#define NWROW (B_SZ / MACRO)       // 64 macro-tiles per dimension
#define NWAVES (NWROW * NWROW)     // 4096 waves total

typedef __attribute__((ext_vector_type(2))) float v2f;  // fp32 A/B frag (16x4): 2 VGPRs
typedef __attribute__((ext_vector_type(8))) float v8f;  // fp32 C/D frag (16x16): 8 VGPRs

// ---------------------------------------------------------------------------
// Kernel 1: sq[i] = sum_k E[i][k]^2   (one wave per row, float4 loads)
// ---------------------------------------------------------------------------
__global__ __launch_bounds__(256) void sq_rows_kernel(const float* __restrict__ E,
                                                      float* __restrict__ sq) {
  const int wave = (blockIdx.x * blockDim.x + threadIdx.x) >> 5;
  const int lane = threadIdx.x & 31;
  const float4* rowp = reinterpret_cast<const float4*>(E + (size_t)wave * D_SZ);
  float s = 0.f;
#pragma unroll
  for (int i = 0; i < D_SZ / (4 * 32); ++i) {  // 6 float4s per lane
    float4 v = rowp[lane + i * 32];
    s += v.x * v.x + v.y * v.y + v.z * v.z + v.w * v.w;
  }
#pragma unroll
  for (int m = 16; m >= 1; m >>= 1) s += __shfl_xor(s, m, 32);
  if (lane == 0) sq[wave] = s;
}

// ---------------------------------------------------------------------------
// Kernel 2: per-wave 64x64 gram macro-tile via V_WMMA_F32_16X16X4_F32, fused
// loss epilogue. Writes one partial sum per wave (deterministic, no atomics).
//
// fp32 A-frag layout (ISA 7.12.2, 16x4 MxK): lanes 0-15 hold M=lane, VGPR0/1 =
// K 0/1; lanes 16-31 hold K 2/3 -> each lane loads 2 consecutive K as one b64.
// B (4x16 KxN) mirrors with N on lanes; since B[k][n] = E[colBase+n][k], the
// B-frag load is identical code against the column block's rows.
// ---------------------------------------------------------------------------
__global__ __launch_bounds__(256) void gram_loss_kernel(const float* __restrict__ E,
                                                        const float* __restrict__ art,
                                                        const float* __restrict__ sq,
                                                        float* __restrict__ partials) {
  const int lane = threadIdx.x & 31;
  const int wave = blockIdx.x * WPB + (threadIdx.x >> 5);
  const int mrow = (wave / NWROW) * MACRO;
  const int mcol = (wave % NWROW) * MACRO;

  const int l15 = lane & 15;
  const int khi = (lane >> 4) * 2;  // this lane covers K = khi, khi+1

  const float* aBase[MT];
  const float* bBase[MT];
#pragma unroll
  for (int t = 0; t < MT; ++t) {
    aBase[t] = E + (size_t)(mrow + t * 16 + l15) * D_SZ + khi;
    bBase[t] = E + (size_t)(mcol + t * 16 + l15) * D_SZ + khi;
  }

  v8f acc[MT][MT] = {};  // zero-init accumulators (128 VGPRs)

  // K loop: 192 steps of 4; 8 b64 loads feed 16 WMMAs per step.
#pragma unroll 2
  for (int k0 = 0; k0 < D_SZ; k0 += 4) {
    v2f a[MT], b[MT];
#pragma unroll
    for (int t = 0; t < MT; ++t) a[t] = *reinterpret_cast<const v2f*>(aBase[t] + k0);
#pragma unroll
    for (int t = 0; t < MT; ++t) b[t] = *reinterpret_cast<const v2f*>(bBase[t] + k0);
#pragma unroll
    for (int i = 0; i < MT; ++i)
#pragma unroll
      for (int j = 0; j < MT; ++j)
        acc[i][j] = __builtin_amdgcn_wmma_f32_16x16x4_f32(
            /*neg_a=*/false, a[i], /*neg_b=*/false, b[j],
            /*c_mod=*/(short)0, acc[i][j], /*reuse_a=*/false, /*reuse_b=*/false);
  }

  // Epilogue: C/D layout -> element (vgpr r, lane): m = r + 8*(lane>=16),
  // n = lane & 15. d2 = sq_i + sq_j - 2*gram; loss += (d2 - art)^2, i != j.
  const int hi8 = (lane >> 4) * 8;
  float loss = 0.f;
#pragma unroll
  for (int i = 0; i < MT; ++i) {
    const int row0 = mrow + i * 16 + hi8;
    float srow[8];
#pragma unroll
    for (int r = 0; r < 8; ++r) srow[r] = sq[row0 + r];
#pragma unroll
    for (int j = 0; j < MT; ++j) {
      const int col = mcol + j * 16 + l15;
      const float scol = sq[col];
      const float* artp = art + (size_t)row0 * B_SZ + col;
#pragma unroll
      for (int r = 0; r < 8; ++r) {
        const float g = acc[i][j][r];
        const float d2 = srow[r] + scol - 2.0f * g;
        // art_dist is 67MB read exactly once -> non-temporal, keep L2 for E.
        const float a = __builtin_nontemporal_load(artp + (size_t)r * B_SZ);
        const float t = d2 - a;
        loss += (row0 + r == col) ? 0.f : t * t;
      }
    }
  }
#pragma unroll
  for (int m = 16; m >= 1; m >>= 1) loss += __shfl_xor(loss, m, 32);
  if (lane == 0) partials[wave] = loss;
}

// ---------------------------------------------------------------------------
// Kernel 3: deterministic fixed-order reduction of 4096 partials -> scalar.
// ---------------------------------------------------------------------------
__global__ __launch_bounds__(256) void reduce_kernel(const float* __restrict__ partials,
                                                     float* __restrict__ out) {
  __shared__ float smem[256];
  const int tid = threadIdx.x;
  float s = 0.f;
  for (int i = tid; i < NWAVES; i += 256) s += partials[i];
  smem[tid] = s;
  __syncthreads();
#pragma unroll
  for (int w = 128; w > 0; w >>= 1) {
    if (tid < w) smem[tid] += smem[tid + w];
    __syncthreads();
  }
  if (tid == 0) {
    const float inv_pairs = 1.0f / ((float)B_SZ * (float)(B_SZ - 1));
    out[0] = smem[0] * inv_pairs;
  }
}

// ---------------------------------------------------------------------------
extern "C" void kernel_launch(void* const* d_in, const int* in_sizes, int n_in,
                              void* d_out, int out_size, void* d_ws, size_t ws_size,
                              hipStream_t stream) {
  const float* E = (const float*)d_in[0];    // [4096, 768] fp32
  const float* art = (const float*)d_in[1];  // [4096, 4096] fp32
  float* out = (float*)d_out;                // scalar fp32

  float* sq = (float*)d_ws;                  // 4096 floats
  float* partials = sq + B_SZ;               // 4096 floats

  sq_rows_kernel<<<B_SZ / WPB, 256, 0, stream>>>(E, sq);
  gram_loss_kernel<<<NWAVES / WPB, 256, 0, stream>>>(E, art, sq, partials);
  reduce_kernel<<<1, 256, 0, stream>>>(partials, out);
}